// DynamicConv2d_57131654971741
// MI455X (gfx1250) — compile-verified
//
#include <hip/hip_runtime.h>

// CDNA5 (gfx1250) dynamic conv as per-batch implicit GEMM on v_wmma_f32_16x16x4_f32,
// with Tensor Data Mover (TDM) staging of weight + input rows into LDS.
// Y[b,o,h,w] = sum_{c,i,j} x[b,c,(h+i-1)%128,(w+j-1)%128] * W[b,c*9+i*3+j,o] + bias[b,o]

typedef float v2f __attribute__((ext_vector_type(2)));
typedef float v8f __attribute__((ext_vector_type(8)));
typedef unsigned int u32x4 __attribute__((ext_vector_type(4)));
typedef int i32x4 __attribute__((ext_vector_type(4)));
typedef int i32x8 __attribute__((ext_vector_type(8)));

#define B_SZ 16
#define C_SZ 64
#define O_SZ 64
#define H_SZ 128
#define W_SZ 128
#define K_SZ 576                         // C * 3 * 3

// LDS strides chosen so 2*stride == 16 (mod 64): the lane0-15 half and the
// lane16-31 half of every WMMA fragment land on disjoint bank ranges.
#define WS_STRIDE 72                     // weight: k-major stride (floats)
#define XS_STRIDE 136                    // input row stride: 128 + 2 halo, padded
#define WS_ELEMS (K_SZ * WS_STRIDE)      // 41472 floats
#define XS_ELEMS (4 * C_SZ * XS_STRIDE)  // 34816 floats (rows h0-1 .. h0+2)
#define LDS_BYTES ((WS_ELEMS + XS_ELEMS) * 4)  // 305152 B  (<= 320KB WGP LDS)

#define HAVE_TDM __has_builtin(__builtin_amdgcn_tensor_load_to_lds)

// ---------------------------------------------------------------------------
// TDM: issue one 2D tile load  global -> LDS  with row padding.
//   gaddr    : global byte address of tile start
//   lds_off  : LDS byte offset of tile start
//   dim0     : tile row length in elements (= tensor_dim0, 4B elements)
//   dim1     : number of rows (= tensor_dim1)
//   stride0  : elements between consecutive rows in global memory
//   pad_iv   : pad_interval code (pad after 2^(code+1) DWORDs)
//   pad_amt  : pad_amount code  (code+1 DWORDs of LDS padding)
// ---------------------------------------------------------------------------
__device__ __attribute__((always_inline)) inline
void tdm_load_2d(unsigned long long gaddr, unsigned int lds_off,
                 unsigned int dim0, unsigned int dim1, unsigned int stride0,
                 unsigned int pad_iv, unsigned int pad_amt)
{
#if HAVE_TDM
    u32x4 g0;
    g0[0] = 1u;                                          // count=1, is_restore=0
    g0[1] = lds_off;                                     // lds_addr (bytes)
    g0[2] = (unsigned int)(gaddr & 0xFFFFFFFFull);       // global_addr[31:0]
    g0[3] = (unsigned int)((gaddr >> 32) & 0x1FFFFFFu)   // global_addr[56:32]
          | (2u << 30);                                  // type=2 ("image")
    i32x8 g1;
    g1[0] = (int)((2u << 16)                             // data_size = 4B
                | (1u << 20)                             // pad_enable
                | (pad_iv  << 22)                        // pad_interval
                | (pad_amt << 25));                      // pad_amount
    g1[1] = (int)(dim0 << 16);                           // tensor_dim0[15:0] @ bits63:48
    g1[2] = (int)((dim0 >> 16) | (dim1 << 16));          // tensor_dim0 hi | tensor_dim1 lo
    g1[3] = (int)((dim1 >> 16) | (dim0 << 16));          // tensor_dim1 hi | tile_dim0
    g1[4] = (int)(dim1 & 0xFFFFu);                       // tile_dim1 (tile_dim2 = 0)
    g1[5] = (int)stride0;                                // tensor_dim0_stride[31:0]
    g1[6] = 0;                                           // stride hi | dim1_stride lo (unused, 2D)
    g1[7] = 0;
    i32x4 z4 = {0, 0, 0, 0};
#if defined(__clang_major__) && (__clang_major__ >= 23)
    i32x8 z8 = {0, 0, 0, 0, 0, 0, 0, 0};
    __builtin_amdgcn_tensor_load_to_lds(g0, g1, z4, z4, z8, 0);
#else
    __builtin_amdgcn_tensor_load_to_lds(g0, g1, z4, z4, 0);
#endif
#endif
}

__global__ __launch_bounds__(256, 1)
void dynconv_wmma_f32(const float* __restrict__ x,
                      const float* __restrict__ wgt,
                      const float* __restrict__ bias,
                      float* __restrict__ out)
{
    extern __shared__ float smem[];
    float* ws = smem;                // ws[k*72 + o]         : per-batch weight
    float* xs = smem + WS_ELEMS;     // xs[(r*64+c)*136 + 1+w]: 4 staged input rows + wrap halo

    const int b   = blockIdx.x >> 6;        // 64 row-pairs per batch
    const int h0  = (blockIdx.x & 63) << 1; // first of two output rows
    const int tid = threadIdx.x;
    const int wave = tid >> 5;

    const float* wsrc = wgt + b * (K_SZ * O_SZ);
    const float* xsrc = x + b * (C_SZ * H_SZ * W_SZ);

#if HAVE_TDM
    // ---- TDM staging: waves 0-3 pull one input row each, wave 4 pulls weight.
    // Dynamic LDS begins right after static LDS (none here) -> groupstaticsize().
    const unsigned int lds_base = __builtin_amdgcn_groupstaticsize();
    if (wave < 4) {
        int g = (h0 - 1 + wave) & (H_SZ - 1);
        // tile: 64 channel-rows of 128 floats, channel-plane stride 128*128;
        // pad 8 DWORDs after every 128 -> LDS stride 136; +4B start = halo col.
        tdm_load_2d((unsigned long long)(const void*)(xsrc + g * W_SZ),
                    lds_base + (unsigned int)(WS_ELEMS + wave * C_SZ * XS_STRIDE) * 4u + 4u,
                    W_SZ, C_SZ, H_SZ * W_SZ,
                    /*pad_iv 2^(6+1)=128*/ 6u, /*pad_amt 8*/ 7u);
        __builtin_amdgcn_s_wait_tensorcnt(0);
    } else if (wave == 4) {
        // tile: 576 k-rows of 64 floats; pad 8 after every 64 -> k-stride 72.
        tdm_load_2d((unsigned long long)(const void*)wsrc,
                    lds_base, O_SZ, K_SZ, O_SZ,
                    /*pad_iv 2^(5+1)=64*/ 5u, /*pad_amt 8*/ 7u);
        __builtin_amdgcn_s_wait_tensorcnt(0);
    }
#else
    // ---- fallback: manual cooperative staging (coalesced) ----------------
    for (int idx = tid; idx < K_SZ * O_SZ; idx += 256) {
        int k = idx >> 6, o = idx & 63;
        ws[k * WS_STRIDE + o] = wsrc[idx];
    }
    for (int idx = tid; idx < 4 * C_SZ * W_SZ; idx += 256) {
        int r   = idx >> 13;
        int rem = idx & 8191;
        int c = rem >> 7, w = rem & 127;
        int g = (h0 - 1 + r) & (H_SZ - 1);
        xs[(r * C_SZ + c) * XS_STRIDE + 1 + w] = xsrc[(c * H_SZ + g) * W_SZ + w];
    }
#endif

    {   // halo columns (wrap in w): 4 rows * 64 channels == 256 threads exactly.
        // Disjoint from all TDM-written LDS locations.
        int r = tid >> 6, c = tid & 63;
        int g = (h0 - 1 + r) & (H_SZ - 1);
        const float* row = xsrc + (c * H_SZ + g) * W_SZ;
        float* dst = xs + (r * C_SZ + c) * XS_STRIDE;
        dst[0]        = row[W_SZ - 1];   // left wrap
        dst[1 + W_SZ] = row[0];          // right wrap
    }
    __syncthreads();

    // ---------------- wave tiling ---------------------------------------------
    const int lane    = tid & 31;
    const int lane15  = lane & 15;
    const int half    = lane >> 4;        // frag K-half: lanes0-15 -> K{0,1}, lanes16-31 -> K{2,3}
    const int row_sel = wave >> 2;        // which of the 2 output rows
    const int w_base  = (wave & 3) << 5;  // 32-pixel span = two 16-wide N tiles

    v8f acc0[4], acc1[4];
    const v8f zacc = {0.f, 0.f, 0.f, 0.f, 0.f, 0.f, 0.f, 0.f};
#pragma unroll
    for (int t = 0; t < 4; ++t) { acc0[t] = zacc; acc1[t] = zacc; }

    // K regrouped as (tap jj,ii) x (channel chunk of 4); k = c*9 + ii*3 + jj.
#pragma unroll
    for (int jj = 0; jj < 3; ++jj) {
#pragma unroll
        for (int ii = 0; ii < 3; ++ii) {
            const float* xrow = xs + ((row_sel + ii) * C_SZ + 2 * half) * XS_STRIDE
                                   + jj + w_base + lane15;
            const float* wrow = ws + (ii * 3 + jj + 18 * half) * WS_STRIDE + lane15;
            for (int c = 0; c < C_SZ; c += 4) {
                // B fragment (4x16 f32): VGPR0 = channel c(+2*half), VGPR1 = +1
                const float* xc = xrow + c * XS_STRIDE;
                v2f bf0, bf1;
                bf0.x = xc[0];   bf0.y = xc[XS_STRIDE];
                bf1.x = xc[16];  bf1.y = xc[XS_STRIDE + 16];
                // A fragments (16x4 f32): VGPR0 = k(+18*half), VGPR1 = k+9 (next channel)
                const float* wc = wrow + c * 9 * WS_STRIDE;
#pragma unroll
                for (int t = 0; t < 4; ++t) {
                    v2f af;
                    af.x = wc[t * 16];
                    af.y = wc[t * 16 + 9 * WS_STRIDE];
                    acc0[t] = __builtin_amdgcn_wmma_f32_16x16x4_f32(
                        false, af, false, bf0, (short)0, acc0[t], false, false);
                    acc1[t] = __builtin_amdgcn_wmma_f32_16x16x4_f32(
                        false, af, false, bf1, (short)0, acc1[t], false, false);
                }
            }
        }
    }

    // ---------------- epilogue: bias + store ----------------------------------
    // C/D layout: VGPR v -> row M=v (lanes 0-15) / M=v+8 (lanes 16-31), N = lane15.
    const int h = h0 + row_sel;
    float* obase = out + ((b * O_SZ) * H_SZ + h) * W_SZ + w_base + lane15;
    const float* bb = bias + b * O_SZ;
#pragma unroll
    for (int t = 0; t < 4; ++t) {
#pragma unroll
        for (int v = 0; v < 8; ++v) {
            int o = t * 16 + v + 8 * half;
            float bo = bb[o];
            obase[o * (H_SZ * W_SZ)]      = acc0[t][v] + bo;
            obase[o * (H_SZ * W_SZ) + 16] = acc1[t][v] + bo;
        }
    }
}

extern "C" void kernel_launch(void* const* d_in, const int* in_sizes, int n_in,
                              void* d_out, int out_size, void* d_ws, size_t ws_size,
                              hipStream_t stream) {
    const float* x    = (const float*)d_in[0];   // [16,64,128,128]
    const float* wgt  = (const float*)d_in[1];   // [16,576,64]
    const float* bias = (const float*)d_in[2];   // [16,64]
    float* out = (float*)d_out;                  // [16,64,128,128]

    dim3 grid(B_SZ * (H_SZ / 2));                // 1024 blocks, each: 64 x (2x128) tile
    dynconv_wmma_f32<<<grid, 256, LDS_BYTES, stream>>>(x, wgt, bias, out);
}